// _FallbackEdgeGraphSAGE_66803921322228
// MI455X (gfx1250) — compile-verified
//
#include <hip/hip_runtime.h>

// ---------------- problem constants (match reference) ----------------
#define N0c 200000
#define N1c 100000
#define N2c 50000
#define INc 64
#define Hc 128
#define EDGE_INc 16
#define NCLSc 8
#define EPSc 1e-5f

typedef __attribute__((ext_vector_type(2))) float v2f;
typedef __attribute__((ext_vector_type(8))) float v8f;

__device__ __forceinline__ v8f wmma_f32(v2f a, v2f b, v8f c) {
  // D = A(16x4) * B(4x16) + C, full f32 (matches reference precision)
  return __builtin_amdgcn_wmma_f32_16x16x4_f32(false, a, false, b, (short)0, c,
                                               false, false);
}

// ---------------- scatter: agg[dst] += feat[src], deg[dst] += 1 -------
__global__ __launch_bounds__(256) void scatter_kernel(
    const float* __restrict__ feat, const int* __restrict__ src,
    const int* __restrict__ dst, float* __restrict__ agg,
    float* __restrict__ deg, int nE, int K) {
  long long w = (long long)blockIdx.x * blockDim.x + threadIdx.x;
  int q = K >> 2;
  long long nwork = (long long)nE * q;
  if (w >= nwork) return;
  int e = (int)(w / q);
  int kq = (int)(w % q) << 2;
  int s = src[e], d = dst[e];
  const float4 val = *(const float4*)(feat + (long long)s * K + kq);
  float* base = agg + (long long)d * K + kq;
  atomicAdd(base + 0, val.x);
  atomicAdd(base + 1, val.y);
  atomicAdd(base + 2, val.z);
  atomicAdd(base + 3, val.w);
  if (kq == 0) atomicAdd(deg + d, 1.0f);
}

// ------- fused SAGE linear: Out = Aself@Ws + (Agg/deg)@Wn + bias ------
// block = 256 threads = 8 waves; block tile = 16 rows x 128 cols
// wave w owns cols [16w, 16w+16); WMMA f32 16x16x4 over K
// epilogue: per-column sum / sumsq for BatchNorm statistics
__global__ __launch_bounds__(256) void sage_gemm_kernel(
    const float* __restrict__ Aself, const float* __restrict__ Agg,
    const float* __restrict__ Deg, const float* __restrict__ Ws,
    const float* __restrict__ Wn, const float* __restrict__ bias,
    float* __restrict__ Out, float* __restrict__ colsum,
    float* __restrict__ colsq, int M, int K) {
  __shared__ alignas(16) float sSelf[16 * 128];
  __shared__ alignas(16) float sMean[16 * 128];
  __shared__ float sSum[128];
  __shared__ float sSq[128];
  int tid = threadIdx.x;
  int tile = blockIdx.x;
  if (tid < 128) {
    sSum[tid] = 0.f;
    sSq[tid] = 0.f;
  }
  // stage A tiles with float4 (rows are 16B aligned; K is 64 or 128)
  int q = K >> 2;           // float4s per row
  int nvec = 16 * q;        // float4s per tile
  for (int idx = tid; idx < nvec; idx += 256) {
    int r = idx / q, kq = (idx - r * q) << 2;
    long long row = (long long)tile * 16 + r;
    float4 sv = *(const float4*)(Aself + row * K + kq);
    float4 av = *(const float4*)(Agg + row * K + kq);
    float rd = 1.0f / fmaxf(Deg[row], 1.0f);
    av.x *= rd; av.y *= rd; av.z *= rd; av.w *= rd;
    *(float4*)(sSelf + r * K + kq) = sv;
    *(float4*)(sMean + r * K + kq) = av;
  }
  __syncthreads();

  int wave = tid >> 5, lane = tid & 31;
  int half = lane >> 4, m = lane & 15;
  int col = (wave << 4) + m;

  float bv = bias[col];
  v8f c;
#pragma unroll
  for (int r = 0; r < 8; ++r) c[r] = bv;

  for (int k0 = 0; k0 < K; k0 += 4) {
    int ka = k0 + 2 * half;  // A VGPR0/1 = K, K+1; lane-half selects K+2,K+3
    v2f a, b;
    a[0] = sSelf[m * K + ka];
    a[1] = sSelf[m * K + ka + 1];
    b[0] = Ws[ka * Hc + col];
    b[1] = Ws[(ka + 1) * Hc + col];
    c = wmma_f32(a, b, c);
    a[0] = sMean[m * K + ka];
    a[1] = sMean[m * K + ka + 1];
    b[0] = Wn[ka * Hc + col];
    b[1] = Wn[(ka + 1) * Hc + col];
    c = wmma_f32(a, b, c);
  }

  float s = 0.f, sq = 0.f;
#pragma unroll
  for (int r = 0; r < 8; ++r) {
    int row = tile * 16 + r + 8 * half;  // C VGPR r: rows r / r+8 per half
    Out[(long long)row * Hc + col] = c[r];
    s += c[r];
    sq += c[r] * c[r];
  }
  atomicAdd(&sSum[col], s);  // LDS atomic: 2 lanes share a column
  atomicAdd(&sSq[col], sq);
  __syncthreads();
  if (tid < 128) {
    atomicAdd(colsum + tid, sSum[tid]);
    atomicAdd(colsq + tid, sSq[tid]);
  }
}

// -------------- BN: turn (sum, sumsq) into (scale, shift) -------------
__global__ void bn_finalize_kernel(const float* __restrict__ colsum,
                                   const float* __restrict__ colsq,
                                   const float* __restrict__ g,
                                   const float* __restrict__ be,
                                   float* __restrict__ scale,
                                   float* __restrict__ shift, float invM) {
  int c = threadIdx.x;  // 128 threads
  float mu = colsum[c] * invM;
  float var = colsq[c] * invM - mu * mu;  // biased var (ddof=0), matches jnp
  float inv = rsqrtf(var + EPSc);
  float sc = g[c] * inv;
  scale[c] = sc;
  shift[c] = be[c] - mu * sc;
}

// ---------- y = relu(scale*x + shift), in place, float4 wide ----------
__global__ __launch_bounds__(256) void bn_relu_kernel(
    float* __restrict__ x, const float* __restrict__ scale,
    const float* __restrict__ shift, long long n4) {
  long long i = (long long)blockIdx.x * blockDim.x + threadIdx.x;
  if (i >= n4) return;
  int c = (int)((i << 2) & (Hc - 1));  // column of first lane element
  float4 v = *(float4*)(x + (i << 2));
  float4 sc = *(const float4*)(scale + c);
  float4 sh = *(const float4*)(shift + c);
  v.x = fmaxf(sc.x * v.x + sh.x, 0.f);
  v.y = fmaxf(sc.y * v.y + sh.y, 0.f);
  v.z = fmaxf(sc.z * v.z + sh.z, 0.f);
  v.w = fmaxf(sc.w * v.w + sh.w, 0.f);
  *(float4*)(x + (i << 2)) = v;
}

// --- edge MLP: out = relu([h[u],h[v],ef] @ W1 + bm1) @ W2 + bm2 -------
// block = 256 threads (8 waves), 16 edges per block; K = 272
__global__ __launch_bounds__(256) void edge_mlp_kernel(
    const float* __restrict__ h2, const int* __restrict__ u,
    const int* __restrict__ v, const float* __restrict__ ef,
    const float* __restrict__ W1, const float* __restrict__ bm1,
    const float* __restrict__ W2, const float* __restrict__ bm2,
    float* __restrict__ out) {
  constexpr int KE = 2 * Hc + EDGE_INc;  // 272 (divisible by 4)
  constexpr int KQ = KE / 4;             // 68 float4s per edge row
  __shared__ alignas(16) float sA[16 * KE];  // 17.4 KB gathered edge features
  __shared__ alignas(16) float sH[16 * Hc];  // hidden activations
  int tid = threadIdx.x, tile = blockIdx.x;

  // float4 gather: h2 rows (128f), ef rows (16f) and sA rows (272f = 1088B)
  // are all 16B aligned.
  for (int idx = tid; idx < 16 * KQ; idx += 256) {
    int r = idx / KQ, k4 = idx - r * KQ;
    int p = tile * 16 + r;
    float4 val;
    if (k4 < Hc / 4)
      val = *(const float4*)(h2 + (long long)u[p] * Hc + (k4 << 2));
    else if (k4 < Hc / 2)
      val = *(const float4*)(h2 + (long long)v[p] * Hc + ((k4 << 2) - Hc));
    else
      val = *(const float4*)(ef + (long long)p * EDGE_INc + ((k4 << 2) - 2 * Hc));
    *(float4*)(sA + r * KE + (k4 << 2)) = val;
  }
  __syncthreads();

  int wave = tid >> 5, lane = tid & 31;
  int half = lane >> 4, m = lane & 15;
  int col = (wave << 4) + m;

  float bv = bm1[col];
  v8f c;
#pragma unroll
  for (int r = 0; r < 8; ++r) c[r] = bv;

  for (int k0 = 0; k0 < KE; k0 += 4) {
    int ka = k0 + 2 * half;
    v2f a, b;
    a[0] = sA[m * KE + ka];
    a[1] = sA[m * KE + ka + 1];
    b[0] = W1[ka * Hc + col];
    b[1] = W1[(ka + 1) * Hc + col];
    c = wmma_f32(a, b, c);
  }
#pragma unroll
  for (int r = 0; r < 8; ++r)
    sH[(r + 8 * half) * Hc + col] = fmaxf(c[r], 0.f);
  __syncthreads();

  // tiny second GEMM: 16 edges x (128 -> 8)
  if (tid < 16 * NCLSc) {
    int e = tid >> 3, cls = tid & 7;
    float acc = bm2[cls];
    for (int j = 0; j < Hc; ++j) acc += sH[e * Hc + j] * W2[j * NCLSc + cls];
    out[((long long)tile * 16 + e) * NCLSc + cls] = acc;
  }
}

// ---------------------------------------------------------------------
extern "C" void kernel_launch(void* const* d_in, const int* in_sizes, int n_in,
                              void* d_out, int out_size, void* d_ws,
                              size_t ws_size, hipStream_t stream) {
  const float* x_nodes = (const float*)d_in[0];
  const int* src0 = (const int*)d_in[1];
  const int* dst0 = (const int*)d_in[2];
  const int* src1 = (const int*)d_in[3];
  const int* dst1 = (const int*)d_in[4];
  const int* u = (const int*)d_in[5];
  const int* vv = (const int*)d_in[6];
  const float* e_feat = (const float*)d_in[7];
  const float* Ws0 = (const float*)d_in[8];
  const float* Wn0 = (const float*)d_in[9];
  const float* b0 = (const float*)d_in[10];
  const float* g0 = (const float*)d_in[11];
  const float* be0 = (const float*)d_in[12];
  const float* Ws1 = (const float*)d_in[13];
  const float* Wn1 = (const float*)d_in[14];
  const float* b1 = (const float*)d_in[15];
  const float* g1 = (const float*)d_in[16];
  const float* be1 = (const float*)d_in[17];
  const float* W1 = (const float*)d_in[18];
  const float* bm1 = (const float*)d_in[19];
  const float* W2 = (const float*)d_in[20];
  const float* bm2 = (const float*)d_in[21];
  float* out = (float*)d_out;

  const int E0 = in_sizes[1];
  const int E1 = in_sizes[3];
  const int EP = in_sizes[5];

  // workspace layout (~129 MB of f32)
  float* agg0 = (float*)d_ws;                      // N1*64
  float* deg0 = agg0 + (size_t)N1c * INc;          // N1
  float* h1 = deg0 + N1c;                          // N1*128
  float* agg1 = h1 + (size_t)N1c * Hc;             // N2*128
  float* deg1 = agg1 + (size_t)N2c * Hc;           // N2
  float* h2 = deg1 + N2c;                          // N2*128
  float* stat = h2 + (size_t)N2c * Hc;             // 512 floats
  float* colsum = stat;
  float* colsq = stat + 128;
  float* scale = stat + 256;
  float* shift = stat + 384;

  // ---------------- layer 0 ----------------
  hipMemsetAsync(agg0, 0, (size_t)N1c * INc * sizeof(float), stream);
  hipMemsetAsync(deg0, 0, (size_t)N1c * sizeof(float), stream);
  hipMemsetAsync(stat, 0, 256 * sizeof(float), stream);
  {
    long long nwork = (long long)E0 * (INc / 4);
    int blocks = (int)((nwork + 255) / 256);
    scatter_kernel<<<blocks, 256, 0, stream>>>(x_nodes, src0, dst0, agg0, deg0,
                                               E0, INc);
  }
  sage_gemm_kernel<<<N1c / 16, 256, 0, stream>>>(
      x_nodes, agg0, deg0, Ws0, Wn0, b0, h1, colsum, colsq, N1c, INc);
  bn_finalize_kernel<<<1, 128, 0, stream>>>(colsum, colsq, g0, be0, scale,
                                            shift, 1.0f / (float)N1c);
  {
    long long n4 = (long long)N1c * Hc / 4;
    bn_relu_kernel<<<(int)((n4 + 255) / 256), 256, 0, stream>>>(h1, scale,
                                                                shift, n4);
  }

  // ---------------- layer 1 ----------------
  hipMemsetAsync(agg1, 0, (size_t)N2c * Hc * sizeof(float), stream);
  hipMemsetAsync(deg1, 0, (size_t)N2c * sizeof(float), stream);
  hipMemsetAsync(stat, 0, 256 * sizeof(float), stream);
  {
    long long nwork = (long long)E1 * (Hc / 4);
    int blocks = (int)((nwork + 255) / 256);
    scatter_kernel<<<blocks, 256, 0, stream>>>(h1, src1, dst1, agg1, deg1, E1,
                                               Hc);
  }
  sage_gemm_kernel<<<N2c / 16, 256, 0, stream>>>(
      h1, agg1, deg1, Ws1, Wn1, b1, h2, colsum, colsq, N2c, Hc);
  bn_finalize_kernel<<<1, 128, 0, stream>>>(colsum, colsq, g1, be1, scale,
                                            shift, 1.0f / (float)N2c);
  {
    long long n4 = (long long)N2c * Hc / 4;
    bn_relu_kernel<<<(int)((n4 + 255) / 256), 256, 0, stream>>>(h2, scale,
                                                                shift, n4);
  }

  // ---------------- edge MLP ----------------
  edge_mlp_kernel<<<EP / 16, 256, 0, stream>>>(h2, u, vv, e_feat, W1, bm1, W2,
                                               bm2, out);
}